// CSA_28647431864328
// MI455X (gfx1250) — compile-verified
//
#include <hip/hip_runtime.h>
#include <hip/hip_bf16.h>

// ---------------------------------------------------------------------------
// Channel self-attention (LayerNorm + grouped-1x1 qkv + softmax attn) for
// B=4, C=256, spatial=32^3 on MI455X (gfx1250, wave32, WMMA).
//
// Pipeline (all deterministic, no float atomics):
//   k_gbsum       : sum(gamma), sum(beta)                       (1 block)
//   k_stats       : per (b,c) sum(x), sum(x^2), sum(x*gamma)    (1024 blocks)
//   k_finalize    : mean, rstd, ysum = rowsum of normalized y   (1024 thr)
//   k_gram        : YY = Y Y^T over needed 96x96 region, split-K partials,
//                   y normalized on the fly during LDS staging, f32 WMMA
//   k_gram_reduce : deterministic sum of split-K partials
//   k_softmax     : per (b,q): logits from YY/ysum closed form, softmax,
//                   fold V-weights into A' (gather, fixed order), attb
//   k_out         : out = x + attb + A' @ Y  (f32 WMMA, y normalized on the
//                   fly during LDS staging)
// ---------------------------------------------------------------------------

typedef __attribute__((ext_vector_type(2))) float v2f;
typedef __attribute__((ext_vector_type(8))) float v8f;

constexpr int BD = 4;
constexpr int CD = 256;
constexpr int SPD = 32 * 32 * 32;   // 32768
constexpr float EPSF = 1e-5f;

// Gram region: rows aq = q/3 in [0,86) ; cols ac = (256+c)/3 in [85,171)
constexpr int GM0 = 0;     // 3 row tiles of 32 -> [0,96)
constexpr int GN0 = 80;    // 3 col tiles of 32 -> [80,176)
constexpr int GTILES = 3;
constexpr int KSPLIT = 16;
constexpr int KCH = SPD / KSPLIT;   // 2048
constexpr int KST = 32;

// ---------------------------------------------------------------------------
__global__ __launch_bounds__(256) void k_gbsum(const float* __restrict__ g,
                                               const float* __restrict__ bt,
                                               float* __restrict__ gsum,
                                               float* __restrict__ bsum) {
  int t = threadIdx.x;
  float a0 = 0.f, a1 = 0.f;
  for (int i = t * 4; i < SPD; i += 256 * 4) {
    float4 gv = *(const float4*)(g + i);
    float4 bv = *(const float4*)(bt + i);
    a0 += gv.x + gv.y + gv.z + gv.w;
    a1 += bv.x + bv.y + bv.z + bv.w;
  }
  __shared__ float r0[256], r1[256];
  r0[t] = a0; r1[t] = a1;
  __syncthreads();
  for (int s = 128; s > 0; s >>= 1) {
    if (t < s) { r0[t] += r0[t + s]; r1[t] += r1[t + s]; }
    __syncthreads();
  }
  if (t == 0) { gsum[0] = r0[0]; bsum[0] = r1[0]; }
}

// ---------------------------------------------------------------------------
__global__ __launch_bounds__(256) void k_stats(const float* __restrict__ x,
                                               const float* __restrict__ g,
                                               float* __restrict__ sx,
                                               float* __restrict__ sxx,
                                               float* __restrict__ sxg) {
  int bc = blockIdx.x;                 // 0..1023
  const float* xr = x + (size_t)bc * SPD;
  int t = threadIdx.x;
  float a0 = 0.f, a1 = 0.f, a2 = 0.f;
  for (int i = t * 4; i < SPD; i += 256 * 4) {
    float4 xv = *(const float4*)(xr + i);
    float4 gv = *(const float4*)(g + i);
    a0 += xv.x + xv.y + xv.z + xv.w;
    a1 += xv.x * xv.x + xv.y * xv.y + xv.z * xv.z + xv.w * xv.w;
    a2 += xv.x * gv.x + xv.y * gv.y + xv.z * gv.z + xv.w * gv.w;
  }
  __shared__ float r0[256], r1[256], r2[256];
  r0[t] = a0; r1[t] = a1; r2[t] = a2;
  __syncthreads();
  for (int s = 128; s > 0; s >>= 1) {
    if (t < s) { r0[t] += r0[t + s]; r1[t] += r1[t + s]; r2[t] += r2[t + s]; }
    __syncthreads();
  }
  if (t == 0) { sx[bc] = r0[0]; sxx[bc] = r1[0]; sxg[bc] = r2[0]; }
}

// ---------------------------------------------------------------------------
__global__ void k_finalize(const float* __restrict__ sx,
                           const float* __restrict__ sxx,
                           const float* __restrict__ sxg,
                           const float* __restrict__ gsum,
                           const float* __restrict__ bsum,
                           float* __restrict__ mean,
                           float* __restrict__ rstd,
                           float* __restrict__ ysum) {
  int i = blockIdx.x * blockDim.x + threadIdx.x;
  if (i >= BD * CD) return;
  float m = sx[i] * (1.f / SPD);
  float var = sxx[i] * (1.f / SPD) - m * m;
  float rs = rsqrtf(var + EPSF);
  mean[i] = m;
  rstd[i] = rs;
  // sum_p y[p] = rs*(sum x*g - m*sum g) + sum beta
  ysum[i] = rs * (sxg[i] - m * gsum[0]) + bsum[0];
}

// ---------------------------------------------------------------------------
// Gram kernel: block tile 32x32, 4 waves (2x2), each wave one 16x16 WMMA tile.
__global__ __launch_bounds__(128) void k_gram(const float* __restrict__ x,
                                              const float* __restrict__ g,
                                              const float* __restrict__ bt,
                                              const float* __restrict__ mean,
                                              const float* __restrict__ rstd,
                                              float* __restrict__ partial) {
  int mt = blockIdx.x, nt = blockIdx.y;
  int b = blockIdx.z / KSPLIT, ks = blockIdx.z % KSPLIT;
  int Mbase = GM0 + mt * 32, Nbase = GN0 + nt * 32;
  const float* xb = x + (size_t)b * CD * SPD;
  const float* mb = mean + b * CD;
  const float* rb = rstd + b * CD;

  __shared__ __align__(16) float As[32][36];
  __shared__ __align__(16) float Bs[32][36];

  int t = threadIdx.x, lane = t & 31, w = t >> 5;
  int wm = w >> 1, wn = w & 1;
  int fr = lane & 15, kh = lane >> 4;

  v8f acc = {};

  int kbeg = ks * KCH;
  for (int k0 = kbeg; k0 < kbeg + KCH; k0 += KST) {
    // stage A rows (channels Mbase..+31) and B rows (channels Nbase..+31),
    // normalized on the fly: (x-mu)*rs*gamma + beta
    for (int i = t; i < 32 * 8; i += 128) {
      int r = i >> 3, c4 = (i & 7) * 4;
      int p = k0 + c4;
      float4 gv = *(const float4*)(g + p);
      float4 bv = *(const float4*)(bt + p);
      {
        int ch = Mbase + r;
        float4 xv = *(const float4*)(xb + (size_t)ch * SPD + p);
        float mu = mb[ch], rs = rb[ch];
        float4 o;
        o.x = (xv.x - mu) * rs * gv.x + bv.x;
        o.y = (xv.y - mu) * rs * gv.y + bv.y;
        o.z = (xv.z - mu) * rs * gv.z + bv.z;
        o.w = (xv.w - mu) * rs * gv.w + bv.w;
        *(float4*)&As[r][c4] = o;
      }
      {
        int ch = Nbase + r;
        float4 xv = *(const float4*)(xb + (size_t)ch * SPD + p);
        float mu = mb[ch], rs = rb[ch];
        float4 o;
        o.x = (xv.x - mu) * rs * gv.x + bv.x;
        o.y = (xv.y - mu) * rs * gv.y + bv.y;
        o.z = (xv.z - mu) * rs * gv.z + bv.z;
        o.w = (xv.w - mu) * rs * gv.w + bv.w;
        *(float4*)&Bs[r][c4] = o;
      }
    }
    __syncthreads();
#pragma unroll
    for (int kk = 0; kk < KST; kk += 4) {
      int kidx = kk + 2 * kh;   // lanes 0-15 -> K {kk,kk+1}; 16-31 -> {kk+2,kk+3}
      v2f a = *(const v2f*)&As[wm * 16 + fr][kidx];
      v2f bfr = *(const v2f*)&Bs[wn * 16 + fr][kidx];
      acc = __builtin_amdgcn_wmma_f32_16x16x4_f32(false, a, false, bfr,
                                                  (short)0, acc, false, false);
    }
    __syncthreads();
  }

  // write 32x32 partial tile for this (b, ks, mt, nt)
  float* pt = partial + (((size_t)b * KSPLIT + ks) * (GTILES * GTILES)
                         + mt * GTILES + nt) * 1024;
#pragma unroll
  for (int r = 0; r < 8; r++) {
    int row = wm * 16 + r + 8 * kh;   // D layout: VGPR r -> row r + 8*(lane>=16)
    int col = wn * 16 + fr;
    pt[row * 32 + col] = acc[r];
  }
}

// ---------------------------------------------------------------------------
__global__ void k_gram_reduce(const float* __restrict__ partial,
                              float* __restrict__ yy) {
  int i = blockIdx.x * 256 + threadIdx.x;   // BD*9*1024 = 36864 total
  if (i >= BD * GTILES * GTILES * 1024) return;
  int b = i / (GTILES * GTILES * 1024);
  int rem = i % (GTILES * GTILES * 1024);
  int tile = rem >> 10, loc = rem & 1023;
  float s = 0.f;
  for (int ks = 0; ks < KSPLIT; ks++)   // fixed order -> deterministic
    s += partial[(((size_t)b * KSPLIT + ks) * (GTILES * GTILES) + tile) * 1024 + loc];
  int mt = tile / GTILES, nt = tile % GTILES;
  int row = GM0 + mt * 32 + (loc >> 5);
  int col = GN0 + nt * 32 + (loc & 31);
  yy[((size_t)b * CD + row) * CD + col] = s;
}

// ---------------------------------------------------------------------------
// Per (b,q): closed-form logits, softmax over c, fold V into A' and attb.
__global__ __launch_bounds__(256) void k_softmax(const float* __restrict__ yy,
                                                 const float* __restrict__ ysum,
                                                 const float* __restrict__ w_qkv,
                                                 const float* __restrict__ b_qkv,
                                                 float* __restrict__ aprime,
                                                 float* __restrict__ attb) {
  int q = blockIdx.x, b = blockIdx.y;
  int c = threadIdx.x;
  const float scale = rsqrtf((float)SPD);

  int aq = q / 3, ac = (CD + c) / 3;
  float wA = w_qkv[q], bA = b_qkv[q];
  float wB = w_qkv[CD + c], bB = b_qkv[CD + c];
  float yyv = yy[((size_t)b * CD + aq) * CD + ac];
  float ysA = ysum[b * CD + aq];
  float ysB = ysum[b * CD + ac];
  float logit = (wA * wB * yyv + wA * bB * ysA + bA * wB * ysB
                 + bA * bB * (float)SPD) * scale;

  __shared__ float sm[256];
  __shared__ float satt[256];

  sm[c] = logit;
  __syncthreads();
  for (int s = 128; s > 0; s >>= 1) {
    if (c < s) sm[c] = fmaxf(sm[c], sm[c + s]);
    __syncthreads();
  }
  float mx = sm[0];
  __syncthreads();

  float e = __expf(logit - mx);
  sm[c] = e;
  __syncthreads();
  for (int s = 128; s > 0; s >>= 1) {
    if (c < s) sm[c] += sm[c + s];
    __syncthreads();
  }
  float inv = 1.f / sm[0];
  __syncthreads();

  float att = e * inv;
  satt[c] = att;
  sm[c] = att * b_qkv[2 * CD + c];   // for attb reduction
  __syncthreads();
  for (int s = 128; s > 0; s >>= 1) {
    if (c < s) sm[c] += sm[c + s];
    __syncthreads();
  }
  if (c == 0) attb[b * CD + q] = sm[0];

  // gather: A'[q][j] = sum over c in {3j-512, 3j-511, 3j-510} of att[c]*wV[c]
  float ap = 0.f;
  int lo = 3 * c - 2 * CD;
#pragma unroll
  for (int d = 0; d < 3; d++) {
    int cc = lo + d;
    if (cc >= 0 && cc < CD) ap += satt[cc] * w_qkv[2 * CD + cc];
  }
  aprime[((size_t)b * CD + q) * CD + c] = ap;
}

// ---------------------------------------------------------------------------
// out = x + attb + A'(256x256) @ Ynorm(256x32768).
// Block tile 64(M) x 128(N), 8 waves as 4(M) x 2(N); each wave 16x64 (4 tiles).
__global__ __launch_bounds__(256) void k_out(const float* __restrict__ x,
                                             const float* __restrict__ g,
                                             const float* __restrict__ bt,
                                             const float* __restrict__ mean,
                                             const float* __restrict__ rstd,
                                             const float* __restrict__ aprime,
                                             const float* __restrict__ attb,
                                             float* __restrict__ out) {
  int Nbase = blockIdx.x * 128;
  int Mbase = blockIdx.y * 64;
  int b = blockIdx.z;
  const float* xb = x + (size_t)b * CD * SPD;
  const float* Ab = aprime + (size_t)b * CD * CD;
  const float* mb = mean + b * CD;
  const float* rb = rstd + b * CD;

  __shared__ __align__(16) float As[64][36];
  __shared__ __align__(16) float Bs[32][136];

  int t = threadIdx.x, lane = t & 31, w = t >> 5;
  int wm = w >> 1;          // 0..3
  int wn = w & 1;           // 0..1
  int fr = lane & 15, kh = lane >> 4;

  v8f zero = {};
  v8f acc[4] = {zero, zero, zero, zero};

  for (int k0 = 0; k0 < CD; k0 += 32) {
    // stage A' chunk: 64 rows x 32 k
    for (int i = t; i < 64 * 8; i += 256) {
      int r = i >> 3, c4 = (i & 7) * 4;
      float4 av = *(const float4*)(Ab + (size_t)(Mbase + r) * CD + k0 + c4);
      *(float4*)&As[r][c4] = av;
    }
    // stage Y chunk: channels k0..k0+31, spatial Nbase..Nbase+127, normalized
    for (int i = t; i < 32 * 32; i += 256) {
      int r = i >> 5, c4 = (i & 31) * 4;
      int ch = k0 + r;
      int p = Nbase + c4;
      float4 xv = *(const float4*)(xb + (size_t)ch * SPD + p);
      float4 gv = *(const float4*)(g + p);
      float4 bv = *(const float4*)(bt + p);
      float mu = mb[ch], rs = rb[ch];
      float4 o;
      o.x = (xv.x - mu) * rs * gv.x + bv.x;
      o.y = (xv.y - mu) * rs * gv.y + bv.y;
      o.z = (xv.z - mu) * rs * gv.z + bv.z;
      o.w = (xv.w - mu) * rs * gv.w + bv.w;
      *(float4*)&Bs[r][c4] = o;
    }
    __syncthreads();
#pragma unroll
    for (int kk = 0; kk < 32; kk += 4) {
      int kidx = kk + 2 * kh;
      v2f a = *(const v2f*)&As[wm * 16 + fr][kidx];
#pragma unroll
      for (int ns = 0; ns < 4; ns++) {
        int ncol = wn * 64 + ns * 16 + fr;
        v2f bf;
        bf.x = Bs[kidx][ncol];
        bf.y = Bs[kidx + 1][ncol];
        acc[ns] = __builtin_amdgcn_wmma_f32_16x16x4_f32(false, a, false, bf,
                                                        (short)0, acc[ns],
                                                        false, false);
      }
    }
    __syncthreads();
  }

  float* ob = out + (size_t)b * CD * SPD;
#pragma unroll
  for (int ns = 0; ns < 4; ns++) {
#pragma unroll
    for (int r = 0; r < 8; r++) {
      int q = Mbase + wm * 16 + r + 8 * kh;
      int p = Nbase + wn * 64 + ns * 16 + fr;
      float xv = xb[(size_t)q * SPD + p];
      ob[(size_t)q * SPD + p] = xv + attb[b * CD + q] + acc[ns][r];
    }
  }
}

// ---------------------------------------------------------------------------
extern "C" void kernel_launch(void* const* d_in, const int* in_sizes, int n_in,
                              void* d_out, int out_size, void* d_ws, size_t ws_size,
                              hipStream_t stream) {
  const float* x     = (const float*)d_in[0];
  const float* gamma = (const float*)d_in[1];
  const float* beta  = (const float*)d_in[2];
  const float* w_qkv = (const float*)d_in[3];
  const float* b_qkv = (const float*)d_in[4];
  float* out = (float*)d_out;

  float* wsf   = (float*)d_ws;
  float* gsum  = wsf + 0;
  float* bsum  = wsf + 1;
  float* sx    = wsf + 16;
  float* sxx   = sx + BD * CD;
  float* sxg   = sxx + BD * CD;
  float* mean  = sxg + BD * CD;
  float* rstd  = mean + BD * CD;
  float* ysum  = rstd + BD * CD;
  float* attb  = ysum + BD * CD;
  float* yy      = wsf + 16384;                    // BD*CD*CD = 262144 floats
  float* aprime  = yy + (size_t)BD * CD * CD;      // 262144 floats
  float* partial = aprime + (size_t)BD * CD * CD;  // BD*KSPLIT*9*1024 floats

  k_gbsum<<<1, 256, 0, stream>>>(gamma, beta, gsum, bsum);
  k_stats<<<BD * CD, 256, 0, stream>>>(x, gamma, sx, sxx, sxg);
  k_finalize<<<(BD * CD + 255) / 256, 256, 0, stream>>>(sx, sxx, sxg, gsum, bsum,
                                                        mean, rstd, ysum);
  k_gram<<<dim3(GTILES, GTILES, BD * KSPLIT), 128, 0, stream>>>(
      x, gamma, beta, mean, rstd, partial);
  k_gram_reduce<<<(BD * GTILES * GTILES * 1024 + 255) / 256, 256, 0, stream>>>(
      partial, yy);
  k_softmax<<<dim3(CD, BD), 256, 0, stream>>>(yy, ysum, w_qkv, b_qkv, aprime, attb);
  k_out<<<dim3(SPD / 128, CD / 64, BD), 256, 0, stream>>>(
      x, gamma, beta, mean, rstd, aprime, attb, out);
}